// SimpleRNN_30562987278660
// MI455X (gfx1250) — compile-verified
//
#include <hip/hip_runtime.h>
#include <math.h>

// Elman RNN, T=2,000,000 sequential steps, H=30, I=2, O=1.
// Single wave32 owns the serial chain. Lane i holds W_hh row i; h is kept
// broadcast in SGPRs via v_readlane so the 30x30 matvec is 30 v_fmac with
// SGPR x VGPR operands. Output projection per 32-step group is done with
// exact-fp32 V_WMMA_F32_16X16X4_F32 chains fed from an LDS h-history.

#define RNN_T   2000000
#define RNN_H   30
#define GROUPS  (RNN_T / 32)   // 62,500 exactly
#define ROWSTRIDE 34           // even stride: b64-aligned LDS rows, no bank conflicts

typedef float v8f __attribute__((ext_vector_type(8)));
typedef float v2f __attribute__((ext_vector_type(2)));

#if __has_builtin(__builtin_amdgcn_wmma_f32_16x16x4_f32)
#define HAVE_WMMA_F32X4 1
#endif

__device__ __forceinline__ float lane_bcast(float v, int lane) {
  return __uint_as_float(__builtin_amdgcn_readlane(__float_as_uint(v), lane));
}

__device__ __forceinline__ float tanh_fast(float a) {
#if __has_builtin(__builtin_amdgcn_tanhf)
  return __builtin_amdgcn_tanhf(a);          // v_tanh_f32 (TRANS)
#elif __has_builtin(__builtin_amdgcn_exp2f) && __has_builtin(__builtin_amdgcn_rcpf)
  // tanh(a) = 1 - 2/(exp(2a)+1);  exp(2a) = exp2(a * 2*log2(e))
  float e = __builtin_amdgcn_exp2f(a * 2.8853900817779268f);
  return 1.0f - 2.0f * __builtin_amdgcn_rcpf(e + 1.0f);
#else
  return tanhf(a);
#endif
}

__global__ __launch_bounds__(32, 1) void rnn_seq_kernel(
    const float* __restrict__ x,    // [T,2]
    const float* __restrict__ Wih,  // [30,2]
    const float* __restrict__ Whh,  // [30,30]
    const float* __restrict__ bih,  // [30]
    const float* __restrict__ bhh,  // [30]
    const float* __restrict__ Wfc,  // [1,30]
    const float* __restrict__ bfc,  // [1]
    float* __restrict__ out)        // [T,1]
{
  __shared__ __attribute__((aligned(16))) float hbuf[32 * ROWSTRIDE];
  const int lane = (int)threadIdx.x;

  // --- per-lane static state: row i of W_hh, fused biases, W_ih row ---
  float w[RNN_H];
  float base = 0.f, wi0 = 0.f, wi1 = 0.f;
  if (lane < RNN_H) {
#pragma unroll
    for (int j = 0; j < RNN_H; ++j) w[j] = Whh[lane * RNN_H + j];
    base = bih[lane] + bhh[lane];
    wi0  = Wih[lane * 2 + 0];
    wi1  = Wih[lane * 2 + 1];
  } else {
#pragma unroll
    for (int j = 0; j < RNN_H; ++j) w[j] = 0.f;
    // base/wi* stay 0 -> lanes 30,31 compute h == tanh(0) == 0 exactly
  }

  // W_fc padded to 32 (uniform scalar loads)
  float wfull[32];
#pragma unroll
  for (int i = 0; i < 32; ++i) wfull[i] = (i < RNN_H) ? Wfc[i] : 0.f;
  const float bias_o = bfc[0];

#if HAVE_WMMA_F32X4
  // A_p[m][k] = wfull[4p+k], replicated over m (loop-invariant).
  // f32 16x4 A layout: VGPR = k&1; lanes 0-15 -> k in {0,1}, lanes 16-31 -> {2,3}.
  v2f apair[8];
#pragma unroll
  for (int p = 0; p < 8; ++p) {
    apair[p].x = (lane < 16) ? wfull[4 * p + 0] : wfull[4 * p + 2];
    apair[p].y = (lane < 16) ? wfull[4 * p + 1] : wfull[4 * p + 3];
  }
  v8f cbias;
#pragma unroll
  for (int i = 0; i < 8; ++i) cbias[i] = bias_o;
#endif

  // --- broadcast h state in SGPRs; h0 = 0 ---
  float hb[RNN_H];
#pragma unroll
  for (int j = 0; j < RNN_H; ++j) hb[j] = 0.f;

  const float2* xv = (const float2*)x;  // lane k of group g holds x[g*32+k][0..1]
  float2 xc = xv[lane];                 // prefetch group 0

  for (int g = 0; g < GROUPS; ++g) {
    float2 xn; xn.x = 0.f; xn.y = 0.f;
    if (g + 1 < GROUPS) xn = xv[(size_t)(g + 1) * 32 + lane];  // prefetch next group

    // ---- 32 sequential recurrence steps ----
#pragma unroll 2
    for (int s = 0; s < 32; ++s) {
      float x0 = lane_bcast(xc.x, s);
      float x1 = lane_bcast(xc.y, s);
      float ac0 = fmaf(wi1, x1, fmaf(wi0, x0, base));
      float ac1 = 0.f, ac2 = 0.f, ac3 = 0.f;
#pragma unroll
      for (int j = 0; j < 28; j += 4) {   // 4 independent FMA chains
        ac0 = fmaf(w[j + 0], hb[j + 0], ac0);
        ac1 = fmaf(w[j + 1], hb[j + 1], ac1);
        ac2 = fmaf(w[j + 2], hb[j + 2], ac2);
        ac3 = fmaf(w[j + 3], hb[j + 3], ac3);
      }
      ac0 = fmaf(w[28], hb[28], ac0);
      ac1 = fmaf(w[29], hb[29], ac1);
      float hn = tanh_fast((ac0 + ac1) + (ac2 + ac3));

      hbuf[s * ROWSTRIDE + lane] = hn;    // fire-and-forget h history (lanes 30,31 write 0)

#pragma unroll
      for (int j = 0; j < RNN_H; ++j)     // rebroadcast h for next step
        hb[j] = lane_bcast(hn, j);
    }

    __syncthreads();  // single-wave WG: effectively s_wait_dscnt 0 before cross-lane LDS reads

    // ---- output projection for this group: out[s] = W_fc . H[s] + b_fc ----
    float res;
#if HAVE_WMMA_F32X4
    {
      // B_p[k][n] = H[n][4p+k]: f32 4x16 B layout -> lane n (+16 for k>=2),
      // VGPR = k&1. Consecutive i -> one ds_load_b64 per p per half.
      const float* r1 = &hbuf[(lane & 15) * ROWSTRIDE + ((lane >> 4) << 1)];
      const float* r2 = r1 + 16 * ROWSTRIDE;
      v8f d1 = cbias, d2 = cbias;
#pragma unroll
      for (int p = 0; p < 8; ++p) {
        v2f b1 = *(const v2f*)(r1 + 4 * p);
        v2f b2 = *(const v2f*)(r2 + 4 * p);
        d1 = __builtin_amdgcn_wmma_f32_16x16x4_f32(false, apair[p], false, b1,
                                                   (short)0, d1, false, false);
        d2 = __builtin_amdgcn_wmma_f32_16x16x4_f32(false, apair[p], false, b2,
                                                   (short)0, d2, false, false);
      }
      // D[m][n] = out_n for all m => VGPR0 holds out_n in both lane halves:
      // d1[0]: lanes 0-15 -> out[0..15]; d2[0]: lanes 16-31 -> out[16..31].
      res = (lane < 16) ? d1[0] : d2[0];
    }
#else
    {
      const float* row = &hbuf[lane * ROWSTRIDE];
      float r0 = bias_o, r1 = 0.f;
#pragma unroll
      for (int i = 0; i < RNN_H; i += 2) {
        r0 = fmaf(row[i + 0], wfull[i + 0], r0);
        r1 = fmaf(row[i + 1], wfull[i + 1], r1);
      }
      res = r0 + r1;
    }
#endif
    out[(size_t)g * 32 + lane] = res;  // coalesced 128B store
    xc = xn;
  }
}

extern "C" void kernel_launch(void* const* d_in, const int* in_sizes, int n_in,
                              void* d_out, int out_size, void* d_ws, size_t ws_size,
                              hipStream_t stream) {
  (void)in_sizes; (void)n_in; (void)out_size; (void)d_ws; (void)ws_size;
  const float* x   = (const float*)d_in[0];
  const float* Wih = (const float*)d_in[1];
  const float* Whh = (const float*)d_in[2];
  const float* bih = (const float*)d_in[3];
  const float* bhh = (const float*)d_in[4];
  const float* Wfc = (const float*)d_in[5];
  const float* bfc = (const float*)d_in[6];
  float* out = (float*)d_out;
  rnn_seq_kernel<<<dim3(1), dim3(32), 0, stream>>>(x, Wih, Whh, bih, bhh, Wfc, bfc, out);
}